// ReallySparseAttention_54631984005782
// MI455X (gfx1250) — compile-verified
//
#include <hip/hip_runtime.h>
#include <hip/hip_bf16.h>
#include <stdint.h>

typedef float v2f __attribute__((ext_vector_type(2)));
typedef float v8f __attribute__((ext_vector_type(8)));
typedef unsigned int u32x4 __attribute__((ext_vector_type(4)));
typedef int i32x4 __attribute__((ext_vector_type(4)));
typedef int i32x8 __attribute__((ext_vector_type(8)));

namespace {
constexpr int B = 2, T = 1024, E = 128, H = 8, KMIX = 8, HID = 64, NPTS = 64;
constexpr int BT = B * T, BH = B * H, EH = E * H;   // EH = 1024
constexpr float SIGMA_BOOST = 2.0f, MIN_SIGMA = 0.01f;
constexpr float QK_SCALE = 0.29730177875068026f;    // 128^-0.25

// workspace layout (units of 4 bytes)
constexpr size_t OFF_MEANS = 0;                                   // BT*K*2 f32
constexpr size_t OFF_SIG   = OFF_MEANS + (size_t)BT * KMIX * 2;   // BT*K f32
constexpr size_t OFF_WTS   = OFF_SIG   + (size_t)BT * KMIX;       // BT*NPTS f32
constexpr size_t OFF_IDX   = OFF_WTS   + (size_t)BT * NPTS;       // BT*NPTS*2 i32
constexpr size_t OFF_Q     = OFF_IDX   + (size_t)BT * NPTS * 2;   // BH*T*E f32
constexpr size_t OFF_KM    = OFF_Q     + (size_t)BH * T * E;
constexpr size_t OFF_V     = OFF_KM    + (size_t)BH * T * E;
constexpr size_t OFF_ATT   = OFF_V     + (size_t)BH * T * E;
}

// ---------------------------------------------------------------- threefry2x32
__device__ __forceinline__ uint32_t rotl32(uint32_t x, int n) {
  return (x << n) | (x >> (32 - n));
}

__device__ __forceinline__ void threefry2x32(uint32_t k0, uint32_t k1,
                                             uint32_t c0, uint32_t c1,
                                             uint32_t& o0, uint32_t& o1) {
  const uint32_t ks2 = 0x1BD11BDAu ^ k0 ^ k1;
  uint32_t x0 = c0 + k0, x1 = c1 + k1;
  const int R0[4] = {13, 15, 26, 6}, R1[4] = {17, 29, 16, 24};
#pragma unroll
  for (int i = 0; i < 4; ++i) { x0 += x1; x1 = rotl32(x1, R0[i]); x1 ^= x0; }
  x0 += k1; x1 += ks2 + 1u;
#pragma unroll
  for (int i = 0; i < 4; ++i) { x0 += x1; x1 = rotl32(x1, R1[i]); x1 ^= x0; }
  x0 += ks2; x1 += k0 + 2u;
#pragma unroll
  for (int i = 0; i < 4; ++i) { x0 += x1; x1 = rotl32(x1, R0[i]); x1 ^= x0; }
  x0 += k0; x1 += k1 + 3u;
#pragma unroll
  for (int i = 0; i < 4; ++i) { x0 += x1; x1 = rotl32(x1, R1[i]); x1 ^= x0; }
  x0 += k1; x1 += ks2 + 4u;
#pragma unroll
  for (int i = 0; i < 4; ++i) { x0 += x1; x1 = rotl32(x1, R0[i]); x1 ^= x0; }
  o0 = x0 + ks2; o1 = x1 + k0 + 5u;
}

// ---------------------------------------------------------------- K1: hypernet
// one 64-thread block per token: params = relu(x@Wp1+bp1)@Wp2+bp2 -> means/sigmas
__global__ void __launch_bounds__(64) hyper_kernel(
    const float* __restrict__ x,
    const float* __restrict__ Wp1, const float* __restrict__ bp1,
    const float* __restrict__ Wp2, const float* __restrict__ bp2,
    float* __restrict__ ws) {
  __shared__ float xrow[E];
  __shared__ float h[HID];
  __shared__ float params[3 * KMIX];
  const int tok = blockIdx.x;       // 0..BT-1
  const int tid = threadIdx.x;      // 0..63

  xrow[tid]      = x[(size_t)tok * E + tid];
  xrow[tid + 64] = x[(size_t)tok * E + tid + 64];
  __syncthreads();

  float acc = bp1[tid];
#pragma unroll 4
  for (int e = 0; e < E; ++e) acc = fmaf(xrow[e], Wp1[e * HID + tid], acc);
  h[tid] = fmaxf(acc, 0.0f);
  __syncthreads();

  if (tid < 3 * KMIX) {
    float p = bp2[tid];
#pragma unroll 4
    for (int j = 0; j < HID; ++j) p = fmaf(h[j], Wp2[j * (3 * KMIX) + tid], p);
    params[tid] = p;
  }
  __syncthreads();

  if (tid < KMIX) {
    const int t = tok & (T - 1);
    float m0 = params[2 * tid]     + (float)t;
    float m1 = params[2 * tid + 1] + (float)t;
    if (m1 > m0) { float tmp = m0; m0 = m1; m1 = tmp; }      // _flip (causal)
    m0 = fminf(fmaxf(m0, 0.f), (float)(T - 1));
    m1 = fminf(fmaxf(m1, 0.f), (float)(T - 1));
    float s  = params[2 * KMIX + tid] + SIGMA_BOOST;
    float sp = (s > 20.f) ? s : log1pf(__expf(s));           // softplus
    ws[OFF_MEANS + ((size_t)tok * KMIX + tid) * 2 + 0] = m0;
    ws[OFF_MEANS + ((size_t)tok * KMIX + tid) * 2 + 1] = m1;
    ws[OFF_SIG + (size_t)tok * KMIX + tid] = sp + MIN_SIGMA;
  }
}

// ---------------------------------------------------------------- K2: points + weights
// one wave per token (block = 128 threads = 4 tokens); each lane owns 2 points
__global__ void __launch_bounds__(128) points_kernel(
    const float* __restrict__ mvalues, float* __restrict__ ws) {
  __shared__ int codes[4][NPTS];
  const int wid  = threadIdx.x >> 5;
  const int lane = threadIdx.x & 31;
  const int tok  = blockIdx.x * 4 + wid;
  int* idxws = (int*)(ws + OFF_IDX);

  float mx[KMIX], my[KMIX], sg[KMIX], mv[KMIX];
#pragma unroll
  for (int k = 0; k < KMIX; ++k) {
    mx[k] = ws[OFF_MEANS + ((size_t)tok * KMIX + k) * 2 + 0];
    my[k] = ws[OFF_MEANS + ((size_t)tok * KMIX + k) * 2 + 1];
    sg[k] = ws[OFF_SIG + (size_t)tok * KMIX + k];
    mv[k] = mvalues[k];
  }

  uint32_t kg0, kg1, kn0, kn1;                 // split(key(1)) -> kg, kn
  threefry2x32(0u, 1u, 0u, 1u, kg0, kg1);
  threefry2x32(0u, 1u, 2u, 3u, kn0, kn1);

  int   i0[2], i1[2];
  float dens[2][KMIX];
#pragma unroll
  for (int half = 0; half < 2; ++half) {
    const int p = lane + 32 * half;            // 0..63
    const int k = p >> 3, j = p & 7;           // component, point-within-component
    const int fl0 = (int)floorf(mx[k]);
    const int fl1 = (int)floorf(my[k]);
    int a, bidx;
    if (j < 4) {                               // floor/ceil corners
      a    = min(max(fl0 + (j >> 1), 0), T - 1);
      bidx = min(max(fl1 + (j & 1), 0), T - 1);
    } else if (j < 6) {                        // global random samples
      const uint32_t g  = (uint32_t)(j - 4);
      const uint32_t fi = (((uint32_t)tok * KMIX + (uint32_t)k) * 2u + g) * 2u;
      uint32_t r0, r1, r2, r3;
      threefry2x32(kg0, kg1, fi,      0x9E3779B9u, r0, r1);
      threefry2x32(kg0, kg1, fi + 1u, 0x9E3779B9u, r2, r3);
      a = (int)(r0 & (uint32_t)(T - 1)); bidx = (int)(r2 & (uint32_t)(T - 1));
    } else {                                   // local samples in 2x2 window
      const uint32_t g  = (uint32_t)(j - 6);
      const int b0 = min(max(fl0 - 1, 0), T - 2);
      const int b1 = min(max(fl1 - 1, 0), T - 2);
      const uint32_t fi = (((uint32_t)tok * KMIX + (uint32_t)k) * 2u + g) * 2u;
      uint32_t r0, r1, r2, r3;
      threefry2x32(kn0, kn1, fi,      0x85EBCA6Bu, r0, r1);
      threefry2x32(kn0, kn1, fi + 1u, 0x85EBCA6Bu, r2, r3);
      a = b0 + (int)(r0 & 1u); bidx = b1 + (int)(r2 & 1u);
    }
    if (bidx > a) { int tmp = a; a = bidx; bidx = tmp; }     // _flip
    i0[half] = a; i1[half] = bidx;
    codes[wid][p] = a * T + bidx;
    const float px = (float)a, py = (float)bidx;
#pragma unroll
    for (int k2 = 0; k2 < KMIX; ++k2) {
      const float dx = (px - mx[k2]) / sg[k2];
      const float dy = (py - my[k2]) / sg[k2];
      dens[half][k2] = __expf(-0.5f * (dx * dx + dy * dy));
    }
  }
  __syncthreads();

  // dedup: zero every all-but-first repeat
#pragma unroll
  for (int half = 0; half < 2; ++half) {
    const int p = lane + 32 * half;
    const int c = codes[wid][p];
    bool dup = false;
    for (int q = 0; q < p; ++q) dup |= (codes[wid][q] == c);
    if (dup)
#pragma unroll
      for (int k2 = 0; k2 < KMIX; ++k2) dens[half][k2] = 0.f;
  }

  // per-component normalization over all 64 points (wave reduction)
  float wsum[KMIX];
#pragma unroll
  for (int k2 = 0; k2 < KMIX; ++k2) {
    float s = dens[0][k2] + dens[1][k2];
#pragma unroll
    for (int off = 16; off >= 1; off >>= 1) s += __shfl_xor(s, off, 32);
    wsum[k2] = s;
  }

#pragma unroll
  for (int half = 0; half < 2; ++half) {
    const int p = lane + 32 * half;
    float w = 0.f;
#pragma unroll
    for (int k2 = 0; k2 < KMIX; ++k2) w += mv[k2] * dens[half][k2] / wsum[k2];
    ws[OFF_WTS + (size_t)tok * NPTS + p] = w;
    idxws[((size_t)tok * NPTS + p) * 2 + 0] = i0[half];
    idxws[((size_t)tok * NPTS + p) * 2 + 1] = i1[half];
  }
}

// ---------------------------------------------------------------- K3: QKV GEMM (WMMA f32)
// block: 256 thr (8 waves), 16 rows x 128 cols per block; grid.z selects Wq/Wk/Wv
__global__ void __launch_bounds__(256) qkv_gemm_kernel(
    const float* __restrict__ x,
    const float* __restrict__ Wq, const float* __restrict__ Wk,
    const float* __restrict__ Wv, float* __restrict__ ws) {
  __shared__ float ldsA[16 * E];             // 16 x 128 tile of x (8 KB)
  const int tid  = threadIdx.x;
  const int wave = tid >> 5, lane = tid & 31;
  const int rowBase = blockIdx.x * 16;
  const int nBase   = blockIdx.y * 128 + wave * 16;
  const float* W = (blockIdx.z == 0) ? Wq : (blockIdx.z == 1) ? Wk : Wv;
  const float scl = (blockIdx.z == 2) ? 1.0f : QK_SCALE;
  float* dst = ws + ((blockIdx.z == 0) ? OFF_Q : (blockIdx.z == 1) ? OFF_KM : OFF_V);

  {                                          // stage A tile
    const int r = tid >> 4;
    const int c = (tid & 15) * 8;
    const float4* src = (const float4*)(x + (size_t)(rowBase + r) * E + c);
    float4 v0 = src[0], v1 = src[1];
    *(float4*)(&ldsA[r * E + c])     = v0;
    *(float4*)(&ldsA[r * E + c + 4]) = v1;
  }
  __syncthreads();

  const int m    = lane & 15;
  const int koff = (lane >> 4) * 2;          // lanes 16..31 carry K+2,K+3
  v8f acc = {};
#pragma unroll 4
  for (int k = 0; k < E; k += 4) {
    v2f a, bf;
    a.x  = ldsA[m * E + k + koff];
    a.y  = ldsA[m * E + k + koff + 1];
    bf.x = W[(size_t)(k + koff)     * EH + nBase + m];
    bf.y = W[(size_t)(k + koff + 1) * EH + nBase + m];
    acc = __builtin_amdgcn_wmma_f32_16x16x4_f32(false, a, false, bf,
                                                (short)0, acc, false, false);
  }

#pragma unroll
  for (int v = 0; v < 8; ++v) {              // epilogue into split (b,H,t,e) layout
    const int row = rowBase + v + (lane >> 4) * 8;
    const int col = nBase + m;
    const int t = row & (T - 1), b = row >> 10;
    const int h = col >> 7, e = col & 127;
    dst[(((size_t)(b * H + h)) * T + t) * E + e] = acc[v] * scl;
  }
}

// ---------------------------------------------------------------- K4a: zero accumulator
__global__ void zero_att_kernel(float* __restrict__ ws) {
  const size_t n = (size_t)BH * T * E;
  for (size_t i = (size_t)blockIdx.x * blockDim.x + threadIdx.x; i < n;
       i += (size_t)gridDim.x * blockDim.x)
    ws[OFF_ATT + i] = 0.f;
}

// ---------------------------------------------------------------- K4: gather dot + scatter
// one thread per point; grid = (T, B*H), block = 64
__global__ void __launch_bounds__(64) attn_scatter_kernel(float* __restrict__ ws) {
  const int t  = blockIdx.x;
  const int bh = blockIdx.y;
  const int b  = bh >> 3;                    // / H
  const int p  = threadIdx.x;                // 0..63
  const int* idxws = (const int*)(ws + OFF_IDX);
  const size_t tokp = (size_t)(b * T + t) * NPTS + p;
  const int i = idxws[tokp * 2 + 0];
  const int j = idxws[tokp * 2 + 1];
  const float w = ws[OFF_WTS + tokp];

  const float* Qr = ws + OFF_Q  + ((size_t)bh * T + i) * E;
  const float* Kr = ws + OFF_KM + ((size_t)bh * T + j) * E;
  const float* Vr = ws + OFF_V  + ((size_t)bh * T + j) * E;
  float*       Or = ws + OFF_ATT + ((size_t)bh * T + i) * E;
  __builtin_prefetch(Vr, 0, 1);              // global_prefetch_b8

  float dot = 0.f;
#pragma unroll 8
  for (int e = 0; e < E; e += 4) {
    const float4 q  = *(const float4*)(Qr + e);
    const float4 kk = *(const float4*)(Kr + e);
    dot = fmaf(q.x, kk.x, fmaf(q.y, kk.y, fmaf(q.z, kk.z, fmaf(q.w, kk.w, dot))));
  }
  const float val = dot * w;
#pragma unroll 4
  for (int e = 0; e < E; e += 4) {
    const float4 vv = *(const float4*)(Vr + e);
    atomicAdd(Or + e + 0, val * vv.x);
    atomicAdd(Or + e + 1, val * vv.y);
    atomicAdd(Or + e + 2, val * vv.z);
    atomicAdd(Or + e + 3, val * vv.w);
  }
}

// ---------------------------------------------------------------- K5: output GEMM (WMMA f32)
// block: 256 thr (8 waves) -> 16 rows x 128 cols (all cols); K = 1024.
// A tile (16 merged rows) is a 3D tile of att viewed as (BH, T, E):
//   tile = (h:8) x (t:16) x (e:128), strides (T*E, E, 1) -> one TDM descriptor.
// TDM fills LDS X-fastest: lds[h*2048 + m*128 + e].
__global__ void __launch_bounds__(256) out_gemm_kernel(
    const float* __restrict__ Wu, const float* __restrict__ bu,
    const float* __restrict__ ws, float* __restrict__ out) {
  __shared__ float ldsA[16 * EH];            // 64 KB (WGP has 320 KB)
  const int tid  = threadIdx.x;
  const int wave = tid >> 5, lane = tid & 31;
  const int rowBase = blockIdx.x * 16;
  const int t0 = rowBase & (T - 1), b = rowBase >> 10;

#if __has_builtin(__builtin_amdgcn_tensor_load_to_lds)
  if (wave == 0) {
    // D# group 0: count=1, lds_addr, 57-bit global tile address, type=2
    const uint64_t gaddr =
        (uint64_t)(uintptr_t)(ws + OFF_ATT + ((size_t)b * H * T + t0) * E);
    u32x4 g0;
    g0.x = 1u;                                        // count=1, user mode
    g0.y = (uint32_t)(uintptr_t)(void*)ldsA;          // LDS byte address
    g0.z = (uint32_t)(gaddr & 0xFFFFFFFFu);           // global_addr[31:0]
    g0.w = (uint32_t)((gaddr >> 32) & 0x01FFFFFFu)    // global_addr[56:32]
           | (2u << 30);                              // type=2 ("image")
    // D# group 1: data_size=4B, dim0=128, dim1=1024, tile=(128,16,8),
    //             stride0=128, stride1=131072
    i32x8 g1;
    g1[0] = (int)(2u << 16);                          // data_size=2 -> 4 bytes
    g1[1] = (int)(128u << 16);                        // tensor_dim0[15:0]=128
    g1[2] = (int)(1024u << 16);                       // dim0 hi=0 | tensor_dim1 lo=1024
    g1[3] = (int)(128u << 16);                        // dim1 hi=0 | tile_dim0=128
    g1[4] = (int)(16u | (8u << 16));                  // tile_dim1=16, tile_dim2=8
    g1[5] = 128;                                      // tensor_dim0_stride[31:0]
    g1[6] = 0;                                        // stride0 hi | stride1 lo16 (131072&0xffff=0)
    g1[7] = 2;                                        // tensor_dim1_stride[47:16] = 131072>>16
    // D# group 2: tensor_dim2=BH, no dim3, no iterate
    i32x4 g2; g2[0] = BH; g2[1] = 0; g2[2] = 0; g2[3] = 0;
    i32x4 g3 = {0, 0, 0, 0};
#if __clang_major__ >= 23
    i32x8 g4 = {};                                    // 6-arg form (therock-10.0 headers)
    __builtin_amdgcn_tensor_load_to_lds(g0, g1, g2, g3, g4, 0);
#else
    __builtin_amdgcn_tensor_load_to_lds(g0, g1, g2, g3, 0);   // 5-arg (ROCm 7.2)
#endif
    __builtin_amdgcn_s_wait_tensorcnt(0);
  }
  __syncthreads();
#else
  // fallback: manual staging, lds[h*2048 + r*128 + e]
  for (int iter = 0; iter < 16; ++iter) {
    const int f   = tid + iter * 256;        // float4 index, 0..4095
    const int r   = f >> 8;
    const int col = (f & 255) << 2;
    const int h = col >> 7, e = col & 127;
    const float4 vld =
        *(const float4*)(ws + OFF_ATT + (((size_t)(b * H + h)) * T + t0 + r) * E + e);
    *(float4*)(&ldsA[h * 2048 + r * 128 + e]) = vld;
  }
  __syncthreads();
#endif

  const int m   = lane & 15;
  const int kof = (lane >> 4) * 2;
  v8f acc = {};
#pragma unroll 4
  for (int k = 0; k < EH; k += 4) {
    const int kk0 = k + kof, kk1 = k + kof + 1;
    v2f a, bf;
    a.x  = ldsA[(kk0 >> 7) * 2048 + m * 128 + (kk0 & 127)];
    a.y  = ldsA[(kk1 >> 7) * 2048 + m * 128 + (kk1 & 127)];
    bf.x = Wu[(size_t)kk0 * E + wave * 16 + m];
    bf.y = Wu[(size_t)kk1 * E + wave * 16 + m];
    acc = __builtin_amdgcn_wmma_f32_16x16x4_f32(false, a, false, bf,
                                                (short)0, acc, false, false);
  }

#pragma unroll
  for (int v = 0; v < 8; ++v) {
    const int row = rowBase + v + (lane >> 4) * 8;
    const int col = wave * 16 + m;
    out[(size_t)row * E + col] = acc[v] + bu[col];
  }
}

// ---------------------------------------------------------------- launch
extern "C" void kernel_launch(void* const* d_in, const int* in_sizes, int n_in,
                              void* d_out, int out_size, void* d_ws, size_t ws_size,
                              hipStream_t stream) {
  const float* x   = (const float*)d_in[0];
  const float* Wq  = (const float*)d_in[1];
  const float* Wk  = (const float*)d_in[2];
  const float* Wv  = (const float*)d_in[3];
  const float* Wu  = (const float*)d_in[4];
  const float* bu  = (const float*)d_in[5];
  const float* Wp1 = (const float*)d_in[6];
  const float* bp1 = (const float*)d_in[7];
  const float* Wp2 = (const float*)d_in[8];
  const float* bp2 = (const float*)d_in[9];
  const float* mv  = (const float*)d_in[10];
  float* out = (float*)d_out;
  float* ws  = (float*)d_ws;

  hyper_kernel<<<BT, 64, 0, stream>>>(x, Wp1, bp1, Wp2, bp2, ws);
  points_kernel<<<BT / 4, 128, 0, stream>>>(mv, ws);
  qkv_gemm_kernel<<<dim3(BT / 16, EH / 128, 3), 256, 0, stream>>>(x, Wq, Wk, Wv, ws);
  zero_att_kernel<<<1024, 256, 0, stream>>>(ws);
  attn_scatter_kernel<<<dim3(T, BH), 64, 0, stream>>>(ws);
  out_gemm_kernel<<<BT / 16, 256, 0, stream>>>(Wu, bu, ws, out);
}